// EdgeEmbeddingBlock_13503377179215
// MI455X (gfx1250) — compile-verified
//
#include <hip/hip_runtime.h>

typedef float v2f __attribute__((ext_vector_type(2)));
typedef float v4f __attribute__((ext_vector_type(4)));
typedef float v8f __attribute__((ext_vector_type(8)));

// out layout per edge: k in [0,128), blocks l=0..3, within block k_l = c*(2l+1)+m
// rw columns: col = l*8 + c  (32 cols), Y index: lbase + m (16 values)

__global__ __launch_bounds__(256) void edge_embed_wmma_kernel(
    const float* __restrict__ vec, const float* __restrict__ len,
    const float* __restrict__ W, float* __restrict__ out,
    int E, int nTiles)
{
    // per-wave LDS slice: rw[16 rows][stride 34] = 544 floats, Y[16][16] = 256 floats
    __shared__ float lds[8 * 800];

    const int tid  = threadIdx.x;
    const int lane = tid & 31;
    const int w    = tid >> 5;
    const int h    = lane >> 4;   // lane half (K split)
    const int el   = lane & 15;   // edge-in-tile (load) / N column (B)

    const int tile  = blockIdx.x * 8 + w;
    const int tilec = (tile < nTiles) ? tile : (nTiles - 1);

    // ---------- load edge data (branchless clamp, keeps EXEC full) ----------
    long e = (long)tilec * 16 + el;
    if (e >= E) e = (long)E - 1;
    const float r  = len[e];
    const float vx = vec[e * 3 + 0];
    const float vy = vec[e * 3 + 1];
    const float vz = vec[e * 3 + 2];
    const float invr = 1.0f / r;

    // ---------- envelope * bessel prefactor ----------
    const float u  = r * 0.2f;                 // r / R_MAX
    const float u2 = u * u, u3 = u2 * u, u6 = u3 * u3, u7 = u6 * u, u8 = u7 * u;
    float env = 1.0f - 28.0f * u6 + 48.0f * u7 - 21.0f * u8;
    env = (u < 1.0f) ? env : 0.0f;
    const float bk = 0.63245553203f * env * invr;   // sqrt(2/R_MAX) * env / r

    // ---------- A matrix: radial tile, 16x8 as two 16x4 halves ----------
    // A layout (16x16x4 f32): VGPR0: lanes0-15 K=0, lanes16-31 K=2; VGPR1: K=1/K=3
    const float PI5 = 0.62831853072f;          // pi / R_MAX
    const int b0 = 2 * h;                      // K base for this lane half
    v2f aLo, aHi;
    aLo.x = bk * __sinf(PI5 * (float)(b0 + 1) * r);  // b = b0
    aLo.y = bk * __sinf(PI5 * (float)(b0 + 2) * r);  // b = b0+1
    aHi.x = bk * __sinf(PI5 * (float)(b0 + 5) * r);  // b = b0+4
    aHi.y = bk * __sinf(PI5 * (float)(b0 + 6) * r);  // b = b0+5

    // ---------- B matrix: Wcat[K=b][col = l*8+c] = W[l*64 + b*8 + c] ----------
#define WIDX(K, col) ((((col) >> 3) << 6) + ((K) << 3) + ((col) & 7))
    const int cN0 = el, cN1 = el + 16;
    v2f bA0, bA1, bB0, bB1;
    bA0.x = W[WIDX(b0,     cN0)]; bA0.y = W[WIDX(b0 + 1, cN0)];
    bA1.x = W[WIDX(b0 + 4, cN0)]; bA1.y = W[WIDX(b0 + 5, cN0)];
    bB0.x = W[WIDX(b0,     cN1)]; bB0.y = W[WIDX(b0 + 1, cN1)];
    bB1.x = W[WIDX(b0 + 4, cN1)]; bB1.y = W[WIDX(b0 + 5, cN1)];
#undef WIDX

    // ---------- rw = radial x Wcat via 4 WMMAs (K accumulation chain) ----------
    v8f acc0 = {}; v8f acc1 = {};
    acc0 = __builtin_amdgcn_wmma_f32_16x16x4_f32(false, aLo, false, bA0, (short)0, acc0, false, false);
    acc0 = __builtin_amdgcn_wmma_f32_16x16x4_f32(false, aHi, false, bA1, (short)0, acc0, false, false);
    acc1 = __builtin_amdgcn_wmma_f32_16x16x4_f32(false, aLo, false, bB0, (short)0, acc1, false, false);
    acc1 = __builtin_amdgcn_wmma_f32_16x16x4_f32(false, aHi, false, bB1, (short)0, acc1, false, false);

    // ---------- spill rw tile to LDS (C layout: VGPR j -> row j + 8h, col = lane&15)
    float* rwl = &lds[w * 800];          // [16][34] padded: conflict-free
    float* yl  = &lds[w * 800 + 544];    // [16][16]
#pragma unroll
    for (int j = 0; j < 8; ++j) {
        const int row = j + 8 * h;
        rwl[row * 34 + el]      = acc0[j];   // cols 0..15
        rwl[row * 34 + 16 + el] = acc1[j];   // cols 16..31
    }

    // ---------- spherical harmonics for edge el (lanes 0-15 only) ----------
    {
        const float nx = vx * invr, ny = vy * invr, nz = vz * invr;
        const float x2 = nx * nx, y2 = ny * ny, z2 = nz * nz;
        if (lane < 16) {
            v4f* yrow = (v4f*)&yl[el * 16];
            v4f ya, yb, yc, yd;
            ya.x = 1.0f;
            ya.y = 1.7320508f * nx; ya.z = 1.7320508f * ny; ya.w = 1.7320508f * nz;
            yb.x = 3.8729833f * nx * nz;
            yb.y = 3.8729833f * nx * ny;
            yb.z = 2.2360680f * (y2 - 0.5f * (x2 + z2));
            yb.w = 3.8729833f * ny * nz;
            yc.x = 1.9364917f * (z2 - x2);
            yc.y = 2.0916501f * nz * (3.0f * x2 - z2);
            yc.z = 10.246951f * nx * ny * nz;
            yc.w = 1.6201852f * nz * (5.0f * y2 - 1.0f);
            yd.x = 1.3228757f * ny * (5.0f * y2 - 3.0f);
            yd.y = 1.6201852f * nx * (5.0f * y2 - 1.0f);
            yd.z = 5.1234753f * ny * (x2 - z2);
            yd.w = 2.0916501f * nx * (x2 - 3.0f * z2);
            yrow[0] = ya; yrow[1] = yb; yrow[2] = yc; yrow[3] = yd;
        }
    }

    __syncthreads();

    // ---------- store stage: lane writes out[e, lane*4 .. lane*4+3] ----------
    int rwi[4], yi[4];
#pragma unroll
    for (int jj = 0; jj < 4; ++jj) {
        const int k = lane * 4 + jj;
        if (k < 8)       { rwi[jj] = k;                  yi[jj] = 0;           }
        else if (k < 32) { const int t = k - 8;  rwi[jj] = 8  + t / 3; yi[jj] = 1 + t % 3; }
        else if (k < 72) { const int t = k - 32; rwi[jj] = 16 + t / 5; yi[jj] = 4 + t % 5; }
        else             { const int t = k - 72; rwi[jj] = 24 + t / 7; yi[jj] = 9 + t % 7; }
    }

    const float scale = 0.35355339059f;   // 1/sqrt(8)
    const bool wvalid = (tile < nTiles);
#pragma unroll 4
    for (int ei = 0; ei < 16; ++ei) {
        const long eo = (long)tile * 16 + ei;
        if (wvalid && eo < (long)E) {
            v4f v;
            v.x = scale * rwl[ei * 34 + rwi[0]] * yl[ei * 16 + yi[0]];
            v.y = scale * rwl[ei * 34 + rwi[1]] * yl[ei * 16 + yi[1]];
            v.z = scale * rwl[ei * 34 + rwi[2]] * yl[ei * 16 + yi[2]];
            v.w = scale * rwl[ei * 34 + rwi[3]] * yl[ei * 16 + yi[3]];
            // streaming 512B/edge per wave, fully contiguous; bypass cache retention
            __builtin_nontemporal_store(v, (v4f*)(out + eo * 128 + lane * 4));
        }
    }
}

extern "C" void kernel_launch(void* const* d_in, const int* in_sizes, int n_in,
                              void* d_out, int out_size, void* d_ws, size_t ws_size,
                              hipStream_t stream) {
    const float* vecs = (const float*)d_in[0];   // [E,3]
    const float* lens = (const float*)d_in[1];   // [E]
    const float* W    = (const float*)d_in[2];   // [4,8,8]
    float* out = (float*)d_out;                  // [E,128]

    const int E = in_sizes[1];
    const int nTiles = (E + 15) / 16;
    const int blocks = (nTiles + 7) / 8;         // 8 waves/block, 1 tile/wave

    hipLaunchKernelGGL(edge_embed_wmma_kernel, dim3(blocks), dim3(256), 0, stream,
                       vecs, lens, W, out, E, nTiles);
}